// Siamese_lstm_89721866813939
// MI455X (gfx1250) — compile-verified
//
#include <hip/hip_runtime.h>
#include <hip/hip_bf16.h>
#include <math.h>

#define HID 15
#define NG  60           // 4*HID gates
#define NGP 64           // padded gate stride (gate g -> col (g/15)*16 + g%15)
#define EMB 300          // embedding dim
#define SEQT 2048        // sequence length
#define KPAD 304         // padded K stride for LDS tiles
#define TPB  8           // time tiles (of 16 steps) per block

typedef __attribute__((ext_vector_type(2))) float v2f;
typedef __attribute__((ext_vector_type(8))) float v8f;

// ---------------------------------------------------------------------------
// Kernel 1: pre-gates GEMM via WMMA fp32 16x16x4.
//   G[seq][t][col(g)] = emb[tok[t]] . W_ih[g,:] + b_ih[g] + b_hh[g]
// Block = 128 threads (4 waves); each block owns 8 consecutive 16-step tiles.
// W_ih is staged zero-padded in LDS once per block -> branch-free inner loop:
//   2x ds_load_b64 + v_wmma_f32_16x16x4_f32 per K-step (75 steps/tile/wave).
// ---------------------------------------------------------------------------
__global__ __launch_bounds__(128)
void pregate_wmma_kernel(const int* __restrict__ tok1,
                         const int* __restrict__ tok2,
                         const float* __restrict__ emb,   // [V,300]
                         const float* __restrict__ W_ih,  // [60,300]
                         const float* __restrict__ b_ih,  // [60]
                         const float* __restrict__ b_hh,  // [60]
                         float* __restrict__ G)           // [2][SEQT][64]
{
    __shared__ float Bl[NGP * KPAD];   // zero-padded W_ih  (77.8 KB)
    __shared__ float Al[16 * KPAD];    // gathered embedding tile (19.5 KB)

    const int seq   = blockIdx.y;
    const int tile0 = blockIdx.x * TPB;
    const int* tok  = (seq == 0) ? tok1 : tok2;
    const int tid   = threadIdx.x;
    const int wave  = tid >> 5;
    const int lane  = tid & 31;

    // ---- stage zero-padded B = W_ih into LDS (once per block) ----
    for (int col = 0; col < NGP; ++col)
        for (int k = tid; k < KPAD; k += 128)
            Bl[col * KPAD + k] =
                (col < NG && k < EMB) ? W_ih[col * EMB + k] : 0.f;

    // ---- per-lane WMMA fragment coordinates (ISA 7.12.2 fp32 layouts) ----
    const int row  = lane & 15;            // A: M (time), B: N (gate col)
    const int kk   = (lane >> 4) * 2;      // K sub-offset inside 4-wide step
    const int bcol = wave * 16 + row;      // B gate column for this lane
    const float* Bp = &Bl[bcol * KPAD];
    const float* Ap = &Al[row * KPAD];

    // ---- output epilogue coordinates ----
    const int n    = lane & 15;
    const int mOff = (lane >> 4) * 8;      // C: VGPR r -> M = r (+8 upper half)
    const int gate = wave * 16 + n;        // logical gate index
    const bool gOk = gate < NG;
    int   ocol = 0;
    float bias = 0.f;
    if (gOk) {
        const int q = gate / 15;           // remap: scan reads cols q*16 + j
        ocol = q * 16 + (gate - q * 15);
        bias = b_ih[gate] + b_hh[gate];
    }

    for (int tt = 0; tt < TPB; ++tt) {
        const int t0 = (tile0 + tt) * 16;

        // gather 16 embedding rows into LDS; wave w loads rows w, w+4, ...
        for (int r = wave; r < 16; r += 4) {
            const float* src = emb + (size_t)tok[t0 + r] * EMB;
            for (int c = lane; c < EMB; c += 32)
                Al[r * KPAD + c] = src[c];
        }
        // prefetch next tile's embedding rows (global_prefetch_b8)
        if (tt + 1 < TPB) {
            const int r   = tid >> 3;           // 0..15
            const int seg = tid & 7;            // 8 x 160B covers 1200B row
            const char* p = (const char*)(emb + (size_t)tok[t0 + 16 + r] * EMB);
            __builtin_prefetch(p + seg * 160, 0, 0);
        }
        __syncthreads();

        v8f acc = {};
#pragma unroll 5
        for (int kb = 0; kb < EMB; kb += 4) {   // 75 WMMA accumulations
            const v2f a = *(const v2f*)&Ap[kb + kk];
            const v2f b = *(const v2f*)&Bp[kb + kk];
            acc = __builtin_amdgcn_wmma_f32_16x16x4_f32(
                      false, a, false, b, (short)0, acc, false, false);
        }

        if (gOk) {
            float* Gout = G + ((size_t)seq * SEQT + t0) * NGP + ocol;
#pragma unroll
            for (int r = 0; r < 8; ++r)
                Gout[(size_t)(mOff + r) * NGP] = acc[r] + bias;
        }
        __syncthreads();   // protect Al before next tile's gather
    }
}

// ---------------------------------------------------------------------------
// Kernel 2: serial LSTM scan, one wave32 per sequence.
// Lane j<15 owns h[j], c[j] and its four W_hh rows in registers; h is
// broadcast with __shfl. Both biases already folded into G.
// Gate columns: i -> G[t][0+j], f -> G[t][16+j], g -> G[t][32+j], o -> G[t][48+j].
// ---------------------------------------------------------------------------
__global__ __launch_bounds__(32)
void lstm_scan_kernel(const float* __restrict__ G,     // [2][SEQT][64]
                      const float* __restrict__ W_hh,  // [60,15]
                      const float* __restrict__ h1_0,
                      const float* __restrict__ c1_0,
                      const float* __restrict__ h2_0,
                      const float* __restrict__ c2_0,
                      float* __restrict__ Vout)        // [2][15]
{
    const int seq = blockIdx.x;
    const int j   = threadIdx.x;
    const bool act = (j < HID);
    const int jj  = act ? j : 0;

    float Wi[HID], Wf[HID], Wg[HID], Wo[HID];
#pragma unroll
    for (int k = 0; k < HID; ++k) {
        Wi[k] = W_hh[(0 * HID + jj) * HID + k];
        Wf[k] = W_hh[(1 * HID + jj) * HID + k];
        Wg[k] = W_hh[(2 * HID + jj) * HID + k];
        Wo[k] = W_hh[(3 * HID + jj) * HID + k];
    }

    float h = act ? ((seq == 0) ? h1_0[jj] : h2_0[jj]) : 0.f;
    float c = act ? ((seq == 0) ? c1_0[jj] : c2_0[jj]) : 0.f;

    const float* Gs = G + (size_t)seq * SEQT * NGP;
    for (int t = 0; t < SEQT; ++t) {
        const float* g = Gs + (size_t)t * NGP;
        // prefetch next timestep's gate row (2 cachelines via 4 lane groups)
        __builtin_prefetch((const char*)(g + NGP) + (j & 3) * 64, 0, 0);
        float ai = g[0 * 16 + jj];
        float af = g[1 * 16 + jj];
        float ag = g[2 * 16 + jj];
        float ao = g[3 * 16 + jj];
#pragma unroll
        for (int k = 0; k < HID; ++k) {
            const float hk = __shfl(h, k, 32);
            ai = fmaf(Wi[k], hk, ai);
            af = fmaf(Wf[k], hk, af);
            ag = fmaf(Wg[k], hk, ag);
            ao = fmaf(Wo[k], hk, ao);
        }
        const float ig = 1.f / (1.f + __expf(-ai));
        const float fg = 1.f / (1.f + __expf(-af));
        const float gg = tanhf(ag);
        const float og = 1.f / (1.f + __expf(-ao));
        c = fg * c + ig * gg;
        h = og * tanhf(c);
    }
    if (act) Vout[seq * HID + j] = h;
}

// ---------------------------------------------------------------------------
// Kernel 3: similarity head. vec[75] -> hidden[25] -> out[2].
// ---------------------------------------------------------------------------
__global__ __launch_bounds__(32)
void head_kernel(const float* __restrict__ V,   // [2][15]
                 const float* __restrict__ W1,  // [25,75]
                 const float* __restrict__ b1,  // [25]
                 const float* __restrict__ W2,  // [2,25]
                 const float* __restrict__ b2,  // [2]
                 float* __restrict__ out)       // [2]
{
    __shared__ float vec[5 * HID];
    __shared__ float hid[25];
    const int t = threadIdx.x;
    if (t < HID) {
        const float a = V[t], b = V[HID + t];
        vec[0 * HID + t] = a;
        vec[1 * HID + t] = fabsf(a - b);
        vec[2 * HID + t] = b;
        vec[3 * HID + t] = a * b;
        vec[4 * HID + t] = 0.5f * (a + b);
    }
    __syncthreads();
    if (t < 25) {
        float acc = b1[t];
        for (int k = 0; k < 5 * HID; ++k)
            acc = fmaf(W1[t * 5 * HID + k], vec[k], acc);
        hid[t] = acc;
    }
    __syncthreads();
    if (t < 2) {
        float acc = b2[t];
        for (int k = 0; k < 25; ++k)
            acc = fmaf(W2[t * 25 + k], hid[k], acc);
        out[t] = acc;
    }
}

// ---------------------------------------------------------------------------
// Launch. Input order (setup_inputs): s1, s2, emb, W_ih, W_hh, b_ih, b_hh,
// W1, b1, W2, b2, h1_0, c1_0, h2_0, c2_0.
// ---------------------------------------------------------------------------
extern "C" void kernel_launch(void* const* d_in, const int* in_sizes, int n_in,
                              void* d_out, int out_size, void* d_ws, size_t ws_size,
                              hipStream_t stream) {
    const int*   s1   = (const int*)  d_in[0];
    const int*   s2   = (const int*)  d_in[1];
    const float* emb  = (const float*)d_in[2];
    const float* W_ih = (const float*)d_in[3];
    const float* W_hh = (const float*)d_in[4];
    const float* b_ih = (const float*)d_in[5];
    const float* b_hh = (const float*)d_in[6];
    const float* W1   = (const float*)d_in[7];
    const float* b1   = (const float*)d_in[8];
    const float* W2   = (const float*)d_in[9];
    const float* b2   = (const float*)d_in[10];
    const float* h1_0 = (const float*)d_in[11];
    const float* c1_0 = (const float*)d_in[12];
    const float* h2_0 = (const float*)d_in[13];
    const float* c2_0 = (const float*)d_in[14];
    float* out = (float*)d_out;

    // Workspace layout: G [2][2048][64] f32 (1 MB), then V [2][15] f32.
    float* G = (float*)d_ws;
    float* V = (float*)((char*)d_ws + (size_t)2 * SEQT * NGP * sizeof(float));

    dim3 gridA(SEQT / 16 / TPB, 2);     // 16 x 2 blocks, 8 tiles each
    pregate_wmma_kernel<<<gridA, 128, 0, stream>>>(s1, s2, emb, W_ih,
                                                   b_ih, b_hh, G);

    lstm_scan_kernel<<<2, 32, 0, stream>>>(G, W_hh,
                                           h1_0, c1_0, h2_0, c2_0, V);

    head_kernel<<<1, 32, 0, stream>>>(V, W1, b1, W2, b2, out);
}